// BasicMOE_12060268167903
// MI455X (gfx1250) — compile-verified
//
#include <hip/hip_runtime.h>

// ---------------------------------------------------------------------------
// BasicMOE on gfx1250 (MI455X, wave32, WMMA).
//
// Reference (with its broadcasting bug) collapses to:
//   w      = softmax(x @ Wg + bg)                    [B,E]
//   z[e,i] = sum_b w[b,e] * x[b,i]                   [E,IN]   (WMMA GEMM)
//   s[e]   = sum_b w[b,e]                            [E]
//   tot[o] = sum_{e,i} z[e,i]*We[e,i,o] + sum_e s[e]*be[e,o]
//   out    = broadcast(tot, [B,OUT])
//
// Cost is one full streaming read of We (1 GiB fp32) -> HBM-bound (~46us at
// 23.3 TB/s). Router GEMM and z-GEMM use V_WMMA_F32_16X16X4_F32; the We
// stream uses b128 loads so vmem issue rate cannot cap bandwidth.
// ---------------------------------------------------------------------------

typedef float v2f __attribute__((ext_vector_type(2)));
typedef float v8f __attribute__((ext_vector_type(8)));

#define BQ   1024
#define INQ  4096
#define OUTQ 4096
#define EQ   16

#define NCHUNK   128
#define KTOT     (EQ * INQ)          // 65536
#define CHUNKLEN (KTOT / NCHUNK)     // 512

// float offsets into workspace
#define W_OFF 0                          // w        : 1024*16
#define S_OFF (W_OFF + BQ * EQ)          // s        : 16 (pad to 64)
#define Z_OFF (S_OFF + 64)               // z        : 16*4096
#define P_OFF (Z_OFF + EQ * INQ)         // partials : 128*4096
#define T_OFF (P_OFF + NCHUNK * OUTQ)    // total    : 4096
// grand total ~2.44 MB of workspace

// ---------------------------------------------------------------------------
// Kernel 1: logits = x @ Wg + bg, then row softmax -> w.
// One wave per 16x16 output tile (16 batch rows x 16 experts).
// A tile layout (fp32 16x4): lanes 0-15 hold M=lane,K={0,1}; lanes 16-31 hold
// M=lane-16,K={2,3}. B tile (4x16) mirrors the K split across lane halves.
// D tile: VGPR r holds M=r (lanes 0-15) and M=r+8 (lanes 16-31), N=lane&15.
// ---------------------------------------------------------------------------
__global__ __launch_bounds__(32) void moe_router_softmax(
    const float* __restrict__ x, const float* __restrict__ Wg,
    const float* __restrict__ bg, float* __restrict__ ws) {
  const int lane = threadIdx.x;
  const int l16  = lane & 15;
  const int half = lane >> 4;
  const int koff = half * 2;
  const int rowBase = blockIdx.x * 16;

  const float* xrow = x + (size_t)(rowBase + l16) * INQ + koff;

  v8f acc = {};
  for (int k0 = 0; k0 < INQ; k0 += 4) {
    v2f a, b;
    a.x = xrow[k0];
    a.y = xrow[k0 + 1];
    b.x = Wg[(k0 + koff) * EQ + l16];
    b.y = Wg[(k0 + koff + 1) * EQ + l16];
    acc = __builtin_amdgcn_wmma_f32_16x16x4_f32(false, a, false, b,
                                                (short)0, acc, false, false);
  }

  const float bias = bg[l16];     // N = l16 = expert index
  float* w = ws + W_OFF;
#pragma unroll
  for (int r = 0; r < 8; ++r) {
    float v = acc[r] + bias;
    // softmax across the 16 experts held by this 16-lane half-group
    float m = v;
    m = fmaxf(m, __shfl_xor(m, 1));
    m = fmaxf(m, __shfl_xor(m, 2));
    m = fmaxf(m, __shfl_xor(m, 4));
    m = fmaxf(m, __shfl_xor(m, 8));
    float e = __expf(v - m);
    float s = e;
    s += __shfl_xor(s, 1);
    s += __shfl_xor(s, 2);
    s += __shfl_xor(s, 4);
    s += __shfl_xor(s, 8);
    const int row = rowBase + r + half * 8;
    w[row * EQ + l16] = e / s;
  }
}

// ---------------------------------------------------------------------------
// Kernel 2: s[e] = sum_b w[b,e]   (tiny, deterministic)
// ---------------------------------------------------------------------------
__global__ __launch_bounds__(32) void moe_expert_sums(float* __restrict__ ws) {
  const int lane = threadIdx.x;
  const int l16  = lane & 15;
  const int half = lane >> 4;
  const float* w = ws + W_OFF;
  float s = 0.f;
  const int b0 = half * (BQ / 2);
  for (int b = b0; b < b0 + BQ / 2; ++b) s += w[b * EQ + l16];
  s += __shfl_xor(s, 16);
  if (half == 0) ws[S_OFF + l16] = s;
}

// ---------------------------------------------------------------------------
// Kernel 3: z = w^T @ x   ([16 x 1024] @ [1024 x 4096], K = batch).
// One wave per 16-column tile of z; 256 WMMAs per wave.
// ---------------------------------------------------------------------------
__global__ __launch_bounds__(32) void moe_weighted_x(
    const float* __restrict__ x, float* __restrict__ ws) {
  const int lane = threadIdx.x;
  const int l16  = lane & 15;
  const int half = lane >> 4;
  const int koff = half * 2;
  const int n0 = blockIdx.x * 16;
  const float* w = ws + W_OFF;

  v8f acc = {};
  for (int k0 = 0; k0 < BQ; k0 += 4) {
    v2f a, b;
    a.x = w[(k0 + koff) * EQ + l16];           // A[m=e=l16][k] = w[b][e]
    a.y = w[(k0 + koff + 1) * EQ + l16];
    b.x = x[(size_t)(k0 + koff) * INQ + n0 + l16];       // B[k][n] = x[b][i]
    b.y = x[(size_t)(k0 + koff + 1) * INQ + n0 + l16];
    acc = __builtin_amdgcn_wmma_f32_16x16x4_f32(false, a, false, b,
                                                (short)0, acc, false, false);
  }

  float* z = ws + Z_OFF;
#pragma unroll
  for (int r = 0; r < 8; ++r) {
    z[(size_t)(r + half * 8) * INQ + n0 + l16] = acc[r];  // z[e][i]
  }
}

// ---------------------------------------------------------------------------
// Kernel 4: partial[c][o] = sum_{k in chunk c} z_flat[k] * We_flat[k][o]
// The HBM-bound stage: streams all 1 GiB of We exactly once. Each thread
// owns 4 consecutive outputs -> global_load_b128, 512 B per wave per load,
// fully coalesced within each k-row. Deterministic chunked partials (no
// float atomics). Prefetch hints for the streaming read.
// ---------------------------------------------------------------------------
__global__ __launch_bounds__(256) void moe_gemv_partials(
    const float* __restrict__ We, float* __restrict__ ws) {
  const int o4 = blockIdx.x * 256 + threadIdx.x;   // float4 index, [0,1024)
  const int c  = blockIdx.y;
  const float* v = ws + Z_OFF;       // z flattened: v[k], k = e*IN + i
  const float4* col = (const float4*)We + o4;      // row stride = OUTQ/4

  float4 acc = {0.f, 0.f, 0.f, 0.f};
  const int kbeg = c * CHUNKLEN;
  for (int k = kbeg; k < kbeg + CHUNKLEN; k += 8) {
    if (k + 16 < KTOT)
      __builtin_prefetch(col + (size_t)(k + 16) * (OUTQ / 4), 0, 0);
#pragma unroll
    for (int u = 0; u < 8; ++u) {
      const float  s = v[k + u];
      const float4 m = col[(size_t)(k + u) * (OUTQ / 4)];
      acc.x = fmaf(s, m.x, acc.x);
      acc.y = fmaf(s, m.y, acc.y);
      acc.z = fmaf(s, m.z, acc.z);
      acc.w = fmaf(s, m.w, acc.w);
    }
  }
  float4* part = (float4*)(ws + P_OFF);
  part[(size_t)c * (OUTQ / 4) + o4] = acc;
}

// ---------------------------------------------------------------------------
// Kernel 5: total[o] = sum_c partial[c][o] + sum_e s[e]*be[e][o]  (b128)
// ---------------------------------------------------------------------------
__global__ __launch_bounds__(256) void moe_reduce_total(
    const float* __restrict__ be, float* __restrict__ ws) {
  const int o4 = blockIdx.x * 256 + threadIdx.x;   // float4 index, [0,1024)
  const float4* part = (const float4*)(ws + P_OFF);
  float4 t = {0.f, 0.f, 0.f, 0.f};
#pragma unroll 4
  for (int c = 0; c < NCHUNK; ++c) {
    const float4 p = part[(size_t)c * (OUTQ / 4) + o4];
    t.x += p.x; t.y += p.y; t.z += p.z; t.w += p.w;
  }
  const float* s = ws + S_OFF;
  const float4* be4 = (const float4*)be;
#pragma unroll
  for (int e = 0; e < EQ; ++e) {
    const float4 b = be4[(size_t)e * (OUTQ / 4) + o4];
    t.x = fmaf(s[e], b.x, t.x);
    t.y = fmaf(s[e], b.y, t.y);
    t.z = fmaf(s[e], b.z, t.z);
    t.w = fmaf(s[e], b.w, t.w);
  }
  ((float4*)(ws + T_OFF))[o4] = t;
}

// ---------------------------------------------------------------------------
// Kernel 6: broadcast total over all 1024 output rows (16 MB of b128 stores).
// ---------------------------------------------------------------------------
__global__ __launch_bounds__(256) void moe_broadcast(
    float* __restrict__ out, const float* __restrict__ ws) {
  const size_t idx = (size_t)blockIdx.x * 256 + threadIdx.x;  // float4 index
  const float4* tot = (const float4*)(ws + T_OFF);
  ((float4*)out)[idx] = tot[idx & (OUTQ / 4 - 1)];
}

// ---------------------------------------------------------------------------
extern "C" void kernel_launch(void* const* d_in, const int* in_sizes, int n_in,
                              void* d_out, int out_size, void* d_ws,
                              size_t ws_size, hipStream_t stream) {
  const float* x  = (const float*)d_in[0];
  const float* Wg = (const float*)d_in[1];
  const float* bg = (const float*)d_in[2];
  const float* We = (const float*)d_in[3];
  const float* be = (const float*)d_in[4];
  float* out = (float*)d_out;
  float* ws  = (float*)d_ws;

  moe_router_softmax<<<BQ / 16, 32, 0, stream>>>(x, Wg, bg, ws);
  moe_expert_sums<<<1, 32, 0, stream>>>(ws);
  moe_weighted_x<<<INQ / 16, 32, 0, stream>>>(x, ws);

  dim3 gPart(OUTQ / (256 * 4), NCHUNK);            // (4, 128)
  moe_gemv_partials<<<gPart, 256, 0, stream>>>(We, ws);

  moe_reduce_total<<<OUTQ / (256 * 4), 256, 0, stream>>>(be, ws);
  moe_broadcast<<<(BQ * OUTQ) / (256 * 4), 256, 0, stream>>>(out, ws);
}